// BatchedGraphSAGEMean1Temporal_40862318854444
// MI455X (gfx1250) — compile-verified
//
#include <hip/hip_runtime.h>
#include <math.h>

// Problem constants (from reference)
#define BB    8
#define NN    512
#define FINC  256
#define FOUTC 256
#define NMAXK 32
#define COLS  1024            // 4*FOUT
#define ROWS  4096            // B*N
#define TILE_M 16
#define AROW  260             // padded LDS row stride (floats): banks (4r+k)%64 conflict-free
#define ATILE (TILE_M*AROW)   // 4160 floats per A tile

typedef float v2f __attribute__((ext_vector_type(2)));
typedef float v8f __attribute__((ext_vector_type(8)));

// ---------------- kernel 1: zero BN stats (2*1024 floats) ----------------
__global__ void __launch_bounds__(256) sage_init_stats(float* __restrict__ stats) {
    int i = blockIdx.x * 256 + threadIdx.x;   // grid = 8 blocks -> 2048 floats
    stats[i] = 0.0f;
}

// ---------------- kernel 2: fused gather-mean + WMMA GEMM + L2norm + ReLU + stat accum ----
__global__ void __launch_bounds__(256, 1)
sage_fused(const float* __restrict__ x,
           const int*   __restrict__ idx1,
           const int*   __restrict__ idx2,
           const int*   __restrict__ idx3,
           const float* __restrict__ Wx, const float* __restrict__ bx,
           const float* __restrict__ Wn, const float* __restrict__ bnb,
           float* __restrict__ out,
           float* __restrict__ gsum, float* __restrict__ gsumsq)
{
    extern __shared__ float smem[];
    float* Alds    = smem;                    // 4 tiles * ATILE floats (x, m1, m2, m3)
    float* biasLds = smem + 4 * ATILE;        // 1024 floats
    float* rowsum  = biasLds + COLS;          // 16 floats
    int*   idxLds  = (int*)(rowsum + 16);     // 3*16*32 ints

    const int t       = threadIdx.x;          // 256 threads = 8 waves (wave32)
    const int rowBase = blockIdx.x * TILE_M;  // 256 blocks
    const int b       = rowBase / NN;         // tile never straddles batches (512 % 16 == 0)
    const int nBase   = rowBase % NN;

    // ---- stage neighbor indices for this tile's 16 nodes (3 sets x 16 rows x 32) ----
    for (int i = t; i < 3 * TILE_M * NMAXK; i += 256) {
        int s   = i / (TILE_M * NMAXK);
        int rem = i - s * (TILE_M * NMAXK);
        int r   = rem >> 5;
        int k   = rem & 31;
        const int* ip = (s == 0) ? idx1 : ((s == 1) ? idx2 : idx3);
        idxLds[i] = ip[(nBase + r) * NMAXK + k];
    }
    // ---- stage combined bias vector [bx, bn, bn, bn] ----
    {
        float bv = bnb[t];
        biasLds[t]       = bx[t];
        biasLds[256 + t] = bv;
        biasLds[512 + t] = bv;
        biasLds[768 + t] = bv;
    }
    __syncthreads();

    // ---- stage A tiles: tile0 = x rows, tiles 1..3 = neighbor means (thread t owns channel t) ----
    const float* xb = x + (size_t)b * NN * FINC;
    for (int r = 0; r < TILE_M; ++r)
        Alds[r * AROW + t] = xb[(nBase + r) * FINC + t];
    for (int s = 0; s < 3; ++s) {
        float* At = Alds + (s + 1) * ATILE;
        const int* il = idxLds + s * TILE_M * NMAXK;
        for (int r = 0; r < TILE_M; ++r) {
            float sum = 0.0f;
            #pragma unroll
            for (int k = 0; k < NMAXK; ++k)
                sum += xb[il[r * NMAXK + k] * FINC + t];
            At[r * AROW + t] = sum * (1.0f / NMAXK);
        }
    }
    __syncthreads();

    // ---- WMMA GEMM: wave w -> segment seg = w/2 (which A tile / weight), half = w&1 (128 cols) ----
    const int lane = t & 31;
    const int wave = t >> 5;
    const int seg  = wave >> 1;       // 0: x@Wx ; 1..3: m_s@Wn
    const int half = wave & 1;        // which 128-column half of the segment
    const int mr   = lane & 15;       // A row / B column within 16x16 tile
    const int kg   = lane >> 4;       // K group (0/1) per f32 16x16x4 layout

    const float* Wseg  = (seg == 0) ? Wx : Wn;
    const float* Arow  = Alds + seg * ATILE + mr * AROW + 2 * kg;
    const float* Wbase = Wseg + (size_t)(half * 128 + mr) * FINC + 2 * kg;

    v8f acc[8] = {};                  // 8 N-tiles of 16 cols -> 128 cols, 16 rows

    for (int k0 = 0; k0 < FINC; k0 += 4) {
        v2f a = *(const v2f*)(Arow + k0);          // ds_load_b64, conflict-free
        #pragma unroll
        for (int nt = 0; nt < 8; ++nt) {
            v2f bf = *(const v2f*)(Wbase + (size_t)nt * 16 * FINC + k0);
            acc[nt] = __builtin_amdgcn_wmma_f32_16x16x4_f32(
                false, a, false, bf, (short)0, acc[nt], false, false);
        }
    }

    // ---- add bias (per output column) ----
    const int colBase = seg * 256 + half * 128;
    #pragma unroll
    for (int nt = 0; nt < 8; ++nt) {
        float bv = biasLds[colBase + nt * 16 + mr];
        #pragma unroll
        for (int r = 0; r < 8; ++r) acc[nt][r] += bv;
    }

    // ---- row L2 norm across all 1024 columns (cross-wave via LDS float atomics) ----
    __syncthreads();
    if (t < 16) rowsum[t] = 0.0f;
    __syncthreads();
    #pragma unroll
    for (int r = 0; r < 8; ++r) {
        float sq = 0.0f;
        #pragma unroll
        for (int nt = 0; nt < 8; ++nt) sq += acc[nt][r] * acc[nt][r];
        atomicAdd(&rowsum[r + 8 * kg], sq);
    }
    __syncthreads();

    float inv[8];
    #pragma unroll
    for (int r = 0; r < 8; ++r)
        inv[r] = 1.0f / fmaxf(sqrtf(rowsum[r + 8 * kg]), 1e-12f);

    // ---- normalize + ReLU, store pre-BN result, accumulate per-channel BN stats ----
    #pragma unroll
    for (int nt = 0; nt < 8; ++nt) {
        int col = colBase + nt * 16 + mr;
        float s1 = 0.0f, s2 = 0.0f;
        #pragma unroll
        for (int r = 0; r < 8; ++r) {
            float v = fmaxf(acc[nt][r] * inv[r], 0.0f);
            s1 += v;
            s2 += v * v;
            out[(size_t)(rowBase + r + 8 * kg) * COLS + col] = v;
        }
        atomicAdd(&gsum[col], s1);
        atomicAdd(&gsumsq[col], s2);
    }
}

// ---------------- kernel 3: apply BatchNorm (training-mode, biased variance) ----------------
__global__ void __launch_bounds__(256) sage_bn_apply(float* __restrict__ out,
                                                     const float* __restrict__ gsum,
                                                     const float* __restrict__ gsumsq,
                                                     const float* __restrict__ gamma,
                                                     const float* __restrict__ beta)
{
    int i = blockIdx.x * 256 + threadIdx.x;       // grid covers ROWS*COLS
    int c = i & (COLS - 1);
    const float invn = 1.0f / (float)ROWS;
    float mu  = gsum[c] * invn;
    float var = gsumsq[c] * invn - mu * mu;
    float v   = out[i];
    out[i] = (v - mu) * rsqrtf(var + 1e-5f) * gamma[c] + beta[c];
}

extern "C" void kernel_launch(void* const* d_in, const int* in_sizes, int n_in,
                              void* d_out, int out_size, void* d_ws, size_t ws_size,
                              hipStream_t stream) {
    (void)in_sizes; (void)n_in; (void)out_size; (void)ws_size;
    const float* x    = (const float*)d_in[0];
    const int*   idx1 = (const int*)  d_in[1];
    const int*   idx2 = (const int*)  d_in[2];
    const int*   idx3 = (const int*)  d_in[3];
    const float* Wx   = (const float*)d_in[4];
    const float* bx   = (const float*)d_in[5];
    const float* Wn   = (const float*)d_in[6];
    const float* bnb  = (const float*)d_in[7];
    const float* gam  = (const float*)d_in[8];
    const float* bet  = (const float*)d_in[9];
    float* out = (float*)d_out;

    float* gsum   = (float*)d_ws;       // 1024 floats
    float* gsumsq = gsum + COLS;        // 1024 floats

    sage_init_stats<<<8, 256, 0, stream>>>(gsum);

    size_t shbytes = (size_t)(4 * ATILE + COLS + 16) * sizeof(float)
                   + (size_t)(3 * TILE_M * NMAXK) * sizeof(int);
    sage_fused<<<ROWS / TILE_M, 256, shbytes, stream>>>(
        x, idx1, idx2, idx3, Wx, bx, Wn, bnb, out, gsum, gsumsq);

    sage_bn_apply<<<(ROWS * COLS) / 256, 256, 0, stream>>>(out, gsum, gsumsq, gam, bet);
}